// DirectionalSAGE_19610820673958
// MI455X (gfx1250) — compile-verified
//
#include <hip/hip_runtime.h>

#define NN 100000
#define NE 1250000
#define D  64
#define K2 128     // concat [agg | x]
#define LDA 132    // padded LDS row stride (128+4) -> conflict-free WMMA fragment reads

typedef __attribute__((ext_vector_type(2))) float v2f;
typedef __attribute__((ext_vector_type(8))) float v8f;

// ---------------- degree: deg[dst] += 1 ----------------
__global__ __launch_bounds__(256)
void deg_kernel(const int* __restrict__ dst, float* __restrict__ deg) {
  int e = blockIdx.x * blockDim.x + threadIdx.x;
  if (e < NE) atomicAdd(&deg[dst[e]], 1.0f);
}

// ---------------- scatter: agg[dst] += x[src] ----------------
// thread = (edge, channel); a wave covers 32 consecutive channels of one edge:
// coalesced 128B load of x[src] row and coalesced f32 atomics into agg[dst].
__global__ __launch_bounds__(256)
void scatter_kernel(const float* __restrict__ x,
                    const int* __restrict__ src,
                    const int* __restrict__ dst,
                    float* __restrict__ agg) {
  int tid = blockIdx.x * blockDim.x + threadIdx.x;   // < 80M, fits in int
  int e = tid >> 6;
  int c = tid & 63;
  if (e < NE) {
    atomicAdd(&agg[dst[e] * D + c], x[src[e] * D + c]);
  }
}

// ---------------- fused combine: relu((agg/deg)@wl^T + bl + x@wr^T) --------
// One K=128 GEMM: A = [agg_mean | x] (16x128 per wave), B[k][n] = Wcat[n][k].
// Block = 128 threads (4 waves); block tile = 64 rows, wave tile = 16 rows x 64 cols.
__global__ __launch_bounds__(128)
void sage_combine_kernel(const float* __restrict__ xin,
                         const float* __restrict__ agg,
                         const float* __restrict__ deg,
                         const float* __restrict__ wl,
                         const float* __restrict__ bl,
                         const float* __restrict__ wr,
                         float* __restrict__ out) {
  __shared__ float sA[64 * LDA];

  const int t = threadIdx.x;
  const int rowBase = blockIdx.x * 64;

  // ---- stage A-tile: 64 rows x 128 cols, fully coalesced (cols 0-63 = agg raw,
  //      cols 64-127 = x). agg is scaled by 1/deg later at fragment-read time.
  {
    const int col = t;                       // blockDim.x == 128
    const bool isAgg = (col < D);
    const float* gsrc = isAgg ? agg : xin;
    const int gc = isAgg ? col : (col - D);
    for (int r = 0; r < 64; ++r) {
      int grow = rowBase + r;
      float v = (grow < NN) ? gsrc[grow * D + gc] : 0.0f;
      sA[r * LDA + col] = v;
    }
  }
  __syncthreads();

  const int lane = t & 31;
  const int w    = t >> 5;          // wave id 0..3
  const int half = lane >> 4;       // 0: K sub 0,1  1: K sub 2,3
  const int l15  = lane & 15;
  const int mloc = (w << 4) + l15;  // A-fragment M row within block tile

  const int growA = rowBase + mloc;
  float dv = (growA < NN) ? deg[growA] : 1.0f;
  float rdeg = 1.0f / fmaxf(dv, 1.0f);

  v8f acc0 = {}, acc1 = {}, acc2 = {}, acc3 = {};
  const float* sArow = &sA[mloc * LDA];

  for (int k0 = 0; k0 < K2; k0 += 4) {
    const int ka = k0 + (half << 1);
    const float sc = (k0 < D) ? rdeg : 1.0f;   // agg half gets mean scaling
    v2f a;
    a.x = sArow[ka] * sc;          // A VGPR0: K = ka   (lanes 0-15 K=0,1 / 16-31 K=2,3)
    a.y = sArow[ka + 1] * sc;      // A VGPR1: K = ka+1

    const float* W = (k0 < D) ? wl : wr;
    const int kk = (k0 < D) ? ka : (ka - D);
    v2f b0, b1, b2, b3;            // B[k][n] = W[n][k]; lane gives N, half gives K pair
    b0.x = W[(l15 +  0) * D + kk];  b0.y = W[(l15 +  0) * D + kk + 1];
    b1.x = W[(l15 + 16) * D + kk];  b1.y = W[(l15 + 16) * D + kk + 1];
    b2.x = W[(l15 + 32) * D + kk];  b2.y = W[(l15 + 32) * D + kk + 1];
    b3.x = W[(l15 + 48) * D + kk];  b3.y = W[(l15 + 48) * D + kk + 1];

    acc0 = __builtin_amdgcn_wmma_f32_16x16x4_f32(false, a, false, b0, (short)0, acc0, false, false);
    acc1 = __builtin_amdgcn_wmma_f32_16x16x4_f32(false, a, false, b1, (short)0, acc1, false, false);
    acc2 = __builtin_amdgcn_wmma_f32_16x16x4_f32(false, a, false, b2, (short)0, acc2, false, false);
    acc3 = __builtin_amdgcn_wmma_f32_16x16x4_f32(false, a, false, b3, (short)0, acc3, false, false);
  }

  // ---- epilogue: bias + relu, C layout: lane -> N, (half,vgpr r) -> M
  const float bb0 = bl[l15];
  const float bb1 = bl[l15 + 16];
  const float bb2 = bl[l15 + 32];
  const float bb3 = bl[l15 + 48];

#pragma unroll
  for (int r = 0; r < 8; ++r) {
    int m = (w << 4) + (half << 3) + r;
    int grow = rowBase + m;
    if (grow < NN) {
      float* orow = &out[grow * D];
      orow[l15]      = fmaxf(acc0[r] + bb0, 0.0f);
      orow[l15 + 16] = fmaxf(acc1[r] + bb1, 0.0f);
      orow[l15 + 32] = fmaxf(acc2[r] + bb2, 0.0f);
      orow[l15 + 48] = fmaxf(acc3[r] + bb3, 0.0f);
    }
  }
}

extern "C" void kernel_launch(void* const* d_in, const int* in_sizes, int n_in,
                              void* d_out, int out_size, void* d_ws, size_t ws_size,
                              hipStream_t stream) {
  (void)in_sizes; (void)n_in; (void)out_size; (void)ws_size;

  const float* x   = (const float*)d_in[0];
  const int*   ei  = (const int*)d_in[1];       // [2, NE]: row0 = src, row1 = dst
  const int*   src = ei;
  const int*   dst = ei + NE;
  const float* wlp[3] = { (const float*)d_in[3], (const float*)d_in[6], (const float*)d_in[9]  };
  const float* blp[3] = { (const float*)d_in[4], (const float*)d_in[7], (const float*)d_in[10] };
  const float* wrp[3] = { (const float*)d_in[5], (const float*)d_in[8], (const float*)d_in[11] };

  // workspace carve-out
  const size_t featBytes = (size_t)NN * D * sizeof(float);     // 25.6 MB
  char* ws = (char*)d_ws;
  float* agg = (float*)ws;  ws += (featBytes + 255) & ~(size_t)255;
  float* deg = (float*)ws;  ws += (((size_t)NN * sizeof(float)) + 255) & ~(size_t)255;
  float* x1  = (float*)ws;  ws += (featBytes + 255) & ~(size_t)255;
  float* x2  = (float*)ws;

  // degree once per launch (edges fixed across layers)
  hipMemsetAsync(deg, 0, (size_t)NN * sizeof(float), stream);
  deg_kernel<<<(NE + 255) / 256, 256, 0, stream>>>(dst, deg);

  const float* xin = x;
  float* outs[3] = { x1, x2, (float*)d_out };
  const int scatterGrid = (NE * 64) / 256;          // 312500, exact
  const int combineGrid = (NN + 63) / 64;           // 1563

  for (int l = 0; l < 3; ++l) {
    hipMemsetAsync(agg, 0, featBytes, stream);
    scatter_kernel<<<scatterGrid, 256, 0, stream>>>(xin, src, dst, agg);
    sage_combine_kernel<<<combineGrid, 128, 0, stream>>>(
        xin, agg, deg, wlp[l], blp[l], wrp[l], outs[l]);
    xin = outs[l];
  }
}